// SSIMLoss_63101659512986
// MI455X (gfx1250) — compile-verified
//
#include <hip/hip_runtime.h>
#include <hip/hip_bf16.h>

// SSIM (11x11 box window) via CDNA5 WMMA f32 banded GEMMs.
// Separable box filter on a 16x16 tile:  Out = Kv(16x26) * ( Q(26x26) * Kh(26x16) )
// implemented with V_WMMA_F32_16X16X4_F32 (7 K=4 chunks per GEMM).

typedef float v2f __attribute__((ext_vector_type(2)));
typedef float v8f __attribute__((ext_vector_type(8)));

#define WIN      11
#define PAD      5          // WIN/2
#define TILE     16
#define HALO     26         // TILE + WIN - 1
#define QSTRIDE  28         // padded row stride (even cols 26,27 zero; 112B rows)
#define QROWS    32         // rows 26..31 zero so K-chunk 6 (cols/rows 24..27) is safe
#define QSZ      (QROWS * QSTRIDE)   // 896 floats
#define TSZ      (QROWS * TILE)      // 512 floats (intermediate T, 32x16)
#define WAVE_LDS (3 * QSZ + TSZ)     // P, Tg, Q, T = 3200 floats per wave
#define WAVES_PB 4
#define IMG      512
#define PLANES   48                  // 16 * 3
#define TILES_PER_PLANE 1024         // 32 * 32
#define NTILES   (PLANES * TILES_PER_PLANE)   // 49152
#define NELEM    12582912.0f                  // 16*3*512*512

#define SSIM_C1  0.0001f    // 0.01^2
#define SSIM_C2  0.0009f    // 0.03^2

// Two banded GEMMs through LDS. All lanes execute (EXEC all-1s at WMMA sites).
// A-frag layout (16x4 f32): lane L: m=L&15; half=L>>4 selects K={0,1} vs {2,3}.
// B-frag layout (4x16 f32): VGPR0 -> K = half?2:0 ; VGPR1 -> K = half?3:1.
// C/D layout: vgpr r holds element [r + 8*half][L&15].
__device__ __forceinline__ v8f box_filter_wmma(const float* __restrict__ Q,
                                               float* __restrict__ T,
                                               const v2f* bh, const v2f* av,
                                               int m, int n, int half) {
    // Phase A: T(32x16) = Q(32x28) x Kh(26x16)   (Kh carries 1/121)
#pragma unroll
    for (int b = 0; b < 2; ++b) {
        const int rb = b * 16;
        v8f acc = {};
#pragma unroll
        for (int kc = 0; kc < 7; ++kc) {
            const int k0 = 4 * kc + (half ? 2 : 0);
            v2f a;
            a.x = Q[(rb + m) * QSTRIDE + k0];
            a.y = Q[(rb + m) * QSTRIDE + k0 + 1];
            acc = __builtin_amdgcn_wmma_f32_16x16x4_f32(
                false, a, false, bh[kc], (short)0, acc, false, false);
        }
#pragma unroll
        for (int r = 0; r < 8; ++r)
            T[(rb + r + 8 * half) * TILE + n] = acc[r];
    }
    // Phase B: Out(16x16) = Kv(16x26) x T(26x16)   (DS ops in-order within wave)
    v8f out = {};
#pragma unroll
    for (int kc = 0; kc < 7; ++kc) {
        const int k0 = 4 * kc + (half ? 2 : 0);
        v2f bf;
        bf.x = T[k0 * TILE + n];
        bf.y = T[(k0 + 1) * TILE + n];
        out = __builtin_amdgcn_wmma_f32_16x16x4_f32(
            false, av[kc], false, bf, (short)0, out, false, false);
    }
    return out;
}

__global__ __launch_bounds__(WAVES_PB * 32)
void ssim_tiles_kernel(const float* __restrict__ pred,
                       const float* __restrict__ target,
                       float* __restrict__ tile_sums) {
    __shared__ float lds[WAVES_PB * WAVE_LDS];

    const int lane = threadIdx.x & 31;
    const int wave = threadIdx.x >> 5;

    float* P  = &lds[wave * WAVE_LDS];
    float* Tg = P + QSZ;
    float* Q  = Tg + QSZ;
    float* Tb = Q + QSZ;

    const int tile  = blockIdx.x * WAVES_PB + wave;
    const int plane = tile >> 10;
    const int tr    = (tile >> 5) & 31;
    const int tc    = tile & 31;
    const int row0  = tr * TILE;
    const int col0  = tc * TILE;

    const float* pBase = pred   + (size_t)plane * (IMG * IMG);
    const float* tBase = target + (size_t)plane * (IMG * IMG);

    // Zero P, Tg, Q (contiguous) so pad rows/cols and OOB halo are 0
    for (int i = lane; i < 3 * QSZ; i += 32) P[i] = 0.0f;

    // Load 26x26 halo with zero padding (matches conv2d SAME zero-pad)
    for (int i = lane; i < HALO * HALO; i += 32) {
        const int r  = i / HALO;
        const int c  = i - r * HALO;
        const int gr = row0 - PAD + r;
        const int gc = col0 - PAD + c;
        if ((unsigned)gr < (unsigned)IMG && (unsigned)gc < (unsigned)IMG) {
            const int g = gr * IMG + gc;
            P[r * QSTRIDE + c]  = pBase[g];
            Tg[r * QSTRIDE + c] = tBase[g];
        }
    }

    const int m = lane & 15;
    const int n = m;              // N index = lane&15 in B/C/D layouts
    const int half = lane >> 4;

    // Constant band-matrix fragments (registers after unroll/SROA)
    v2f bh[7], av[7];
    const float inv_w2 = 1.0f / (float)(WIN * WIN);
#pragma unroll
    for (int kc = 0; kc < 7; ++kc) {
        const int c0 = 4 * kc + (half ? 2 : 0);
        // Kh[c][j] = inv121 iff 0 <= c-j <= 10   (j = n)
        bh[kc].x = ((unsigned)(c0 - n)     <= 10u) ? inv_w2 : 0.0f;
        bh[kc].y = ((unsigned)(c0 + 1 - n) <= 10u) ? inv_w2 : 0.0f;
        // Kv[i][c] = 1 iff 0 <= c-i <= 10        (i = m)
        av[kc].x = ((unsigned)(c0 - m)     <= 10u) ? 1.0f : 0.0f;
        av[kc].y = ((unsigned)(c0 + 1 - m) <= 10u) ? 1.0f : 0.0f;
    }

    // Five filtered quantities: mu1, mu2, E[p^2], E[t^2], E[p*t]
    const v8f Fp = box_filter_wmma(P,  Tb, bh, av, m, n, half);
    const v8f Ft = box_filter_wmma(Tg, Tb, bh, av, m, n, half);

    for (int i = lane; i < HALO * HALO; i += 32) {
        const int r = i / HALO; const int o = r * QSTRIDE + (i - r * HALO);
        Q[o] = P[o] * P[o];
    }
    const v8f Fpp = box_filter_wmma(Q, Tb, bh, av, m, n, half);

    for (int i = lane; i < HALO * HALO; i += 32) {
        const int r = i / HALO; const int o = r * QSTRIDE + (i - r * HALO);
        Q[o] = Tg[o] * Tg[o];
    }
    const v8f Ftt = box_filter_wmma(Q, Tb, bh, av, m, n, half);

    for (int i = lane; i < HALO * HALO; i += 32) {
        const int r = i / HALO; const int o = r * QSTRIDE + (i - r * HALO);
        Q[o] = P[o] * Tg[o];
    }
    const v8f Fpt = box_filter_wmma(Q, Tb, bh, av, m, n, half);

    // Per-element SSIM, accumulate this lane's 8 elements
    float s = 0.0f;
#pragma unroll
    for (int r = 0; r < 8; ++r) {
        const float mu1 = Fp[r],  mu2 = Ft[r];
        const float m11 = mu1 * mu1, m22 = mu2 * mu2, m12 = mu1 * mu2;
        const float s11 = Fpp[r] - m11;
        const float s22 = Ftt[r] - m22;
        const float s12 = Fpt[r] - m12;
        s += ((2.0f * m12 + SSIM_C1) * (2.0f * s12 + SSIM_C2)) /
             ((m11 + m22 + SSIM_C1) * (s11 + s22 + SSIM_C2));
    }

    // wave32 butterfly reduction
#pragma unroll
    for (int off = 16; off > 0; off >>= 1)
        s += __shfl_xor(s, off, 32);

    if (lane == 0) tile_sums[tile] = s;
}

// Deterministic final reduction (fixed tree order) -> mean
__global__ __launch_bounds__(256)
void ssim_reduce_kernel(const float* __restrict__ tile_sums, float* __restrict__ out) {
    __shared__ float buf[256];
    float s = 0.0f;
    for (int i = threadIdx.x; i < NTILES; i += 256) s += tile_sums[i];
    buf[threadIdx.x] = s;
    __syncthreads();
#pragma unroll
    for (int st = 128; st > 0; st >>= 1) {
        if ((int)threadIdx.x < st) buf[threadIdx.x] += buf[threadIdx.x + st];
        __syncthreads();
    }
    if (threadIdx.x == 0) out[0] = buf[0] * (1.0f / NELEM);
}

extern "C" void kernel_launch(void* const* d_in, const int* in_sizes, int n_in,
                              void* d_out, int out_size, void* d_ws, size_t ws_size,
                              hipStream_t stream) {
    const float* pred   = (const float*)d_in[0];
    const float* target = (const float*)d_in[1];
    float* tile_sums    = (float*)d_ws;          // NTILES floats = 192 KB
    float* out          = (float*)d_out;

    const int blocks = NTILES / WAVES_PB;        // 12288 blocks x 128 threads
    ssim_tiles_kernel<<<blocks, WAVES_PB * 32, 0, stream>>>(pred, target, tile_sums);
    ssim_reduce_kernel<<<1, 256, 0, stream>>>(tile_sums, out);
}